// SimpleDeepModel_39290360824676
// MI455X (gfx1250) — compile-verified
//
#include <hip/hip_runtime.h>
#include <hip/hip_bf16.h>

typedef __attribute__((ext_vector_type(2))) float v2f;
typedef __attribute__((ext_vector_type(8))) float v8f;
typedef int i32x4 __attribute__((vector_size(16)));

#define EMBED_DIM   300
#define HIDDEN      500
#define GATES       (4 * HIDDEN)   // 2000
#define SEQ_LEN     2048
#define NUM_CLASSES 20

// recurrence partitioning: 100 blocks x 5 hidden units = 500
#define NBLK 100
#define UPB  5
#define ROWS (4 * UPB)             // 20 gate rows per block

// ---------------------------------------------------------------------------
// CDNA5 async global->LDS helpers (ASYNCcnt-tracked, no VGPR staging).
// Builtins are typed: b32 takes (global int*, lds int*, imm, imm),
// b128 takes (global i32x4*, lds i32x4*, imm, imm).
// ---------------------------------------------------------------------------
__device__ __forceinline__ void async_g2l_b32(const void* g, void* l) {
#if __has_builtin(__builtin_amdgcn_global_load_async_to_lds_b32)
    __builtin_amdgcn_global_load_async_to_lds_b32(
        (__attribute__((address_space(1))) int*)g,
        (__attribute__((address_space(3))) int*)l, 0, 0);
#else
    *(float*)l = *(const float*)g;
#endif
}

__device__ __forceinline__ void async_g2l_b128(const void* g, void* l) {
#if __has_builtin(__builtin_amdgcn_global_load_async_to_lds_b128)
    __builtin_amdgcn_global_load_async_to_lds_b128(
        (__attribute__((address_space(1))) i32x4*)g,
        (__attribute__((address_space(3))) i32x4*)l, 0, 0);
#else
    *(float4*)l = *(const float4*)g;
#endif
}

__device__ __forceinline__ void wait_async0() {
#if __has_builtin(__builtin_amdgcn_s_wait_asynccnt)
    __builtin_amdgcn_s_wait_asynccnt(0);
#elif defined(__gfx1250__)
    asm volatile("s_wait_asynccnt 0x0" ::: "memory");
#endif
}

// ---------------------------------------------------------------------------
// init: zero the grid-barrier counters (workspace is poisoned with 0xAA)
// ---------------------------------------------------------------------------
__global__ void init_ctr_kernel(unsigned* ctr) {
    if (threadIdx.x < 2) ctr[threadIdx.x] = 0u;
}

// ---------------------------------------------------------------------------
// embedding gather: X[t, :] = emb[seq[t], :]
// ---------------------------------------------------------------------------
__global__ void embed_kernel(const int* __restrict__ seq,
                             const float* __restrict__ emb,
                             float* __restrict__ X) {
    const int t = blockIdx.x;
    const float* src = emb + (long)seq[t] * EMBED_DIM;
    float* dst = X + (long)t * EMBED_DIM;
    for (int j = threadIdx.x; j < EMBED_DIM; j += blockDim.x)
        dst[j] = src[j];
}

// ---------------------------------------------------------------------------
// fp32 WMMA GEMM:  Y[M,N] = X[M,K] @ W[N,K]^T + (b1 + b2)   (ldc == N)
// One wave per block; each block owns a 64(M) x 16(N) tile: 4 WMMAs per
// k-step sharing one B fragment.  Register double-buffering: fragments for
// k+4 are fetched before the k-step WMMAs issue, so the compiler can wait on
// loadcnt<=5 instead of a dead-stop s_wait_loadcnt 0x0 per WMMA.
// Fragment layouts per CDNA5 ISA 7.12.2 (32-bit A 16x4 / B 4x16 / C 16x16).
// ---------------------------------------------------------------------------
__global__ void __launch_bounds__(32)
inproj_wmma_kernel(const float* __restrict__ X, int lda,
                   const float* __restrict__ W,   // [N, K] row-major
                   const float* __restrict__ b1,
                   const float* __restrict__ b2,
                   float* __restrict__ Y,
                   int N, int K) {
    const int lane = threadIdx.x & 31;
    const int n0   = blockIdx.x * 16;
    const int m0   = blockIdx.y * 64;
    const int half = lane >> 4;       // 0 = lanes 0-15, 1 = lanes 16-31
    const int kb   = half << 1;       // K sub-offset within 4-wide step: 0 or 2
    const int l15  = lane & 15;

    const float bias = b1[n0 + l15] + b2[n0 + l15];
    v8f acc0, acc1, acc2, acc3;
    for (int v = 0; v < 8; ++v) { acc0[v] = bias; acc1[v] = bias; acc2[v] = bias; acc3[v] = bias; }

    const float* wp = W + (long)(n0 + l15) * K + kb;   // B(k,n) = W[n][k]
    const float* x0 = X + (long)(m0 +  0 + l15) * lda + kb;
    const float* x1 = X + (long)(m0 + 16 + l15) * lda + kb;
    const float* x2 = X + (long)(m0 + 32 + l15) * lda + kb;
    const float* x3 = X + (long)(m0 + 48 + l15) * lda + kb;

    // prologue: fragments for k = 0
    v2f b  = *(const v2f*)(wp);
    v2f a0 = *(const v2f*)(x0);
    v2f a1 = *(const v2f*)(x1);
    v2f a2 = *(const v2f*)(x2);
    v2f a3 = *(const v2f*)(x3);

    for (int k = 4; k < K; k += 4) {
        // prefetch next k-step (loads issued before the WMMAs below)
        v2f bn  = *(const v2f*)(wp + k);
        v2f a0n = *(const v2f*)(x0 + k);
        v2f a1n = *(const v2f*)(x1 + k);
        v2f a2n = *(const v2f*)(x2 + k);
        v2f a3n = *(const v2f*)(x3 + k);

        acc0 = __builtin_amdgcn_wmma_f32_16x16x4_f32(false, a0, false, b, (short)0, acc0, false, false);
        acc1 = __builtin_amdgcn_wmma_f32_16x16x4_f32(false, a1, false, b, (short)0, acc1, false, false);
        acc2 = __builtin_amdgcn_wmma_f32_16x16x4_f32(false, a2, false, b, (short)0, acc2, false, false);
        acc3 = __builtin_amdgcn_wmma_f32_16x16x4_f32(false, a3, false, b, (short)0, acc3, false, false);

        b = bn; a0 = a0n; a1 = a1n; a2 = a2n; a3 = a3n;
    }
    // epilogue: last k-step
    acc0 = __builtin_amdgcn_wmma_f32_16x16x4_f32(false, a0, false, b, (short)0, acc0, false, false);
    acc1 = __builtin_amdgcn_wmma_f32_16x16x4_f32(false, a1, false, b, (short)0, acc1, false, false);
    acc2 = __builtin_amdgcn_wmma_f32_16x16x4_f32(false, a2, false, b, (short)0, acc2, false, false);
    acc3 = __builtin_amdgcn_wmma_f32_16x16x4_f32(false, a3, false, b, (short)0, acc3, false, false);

    const int mrb = half * 8;   // C layout: VGPR v holds rows v (lanes 0-15) / v+8 (lanes 16-31)
    for (int v = 0; v < 8; ++v) {
        Y[(long)(m0 +  0 + mrb + v) * N + n0 + l15] = acc0[v];
        Y[(long)(m0 + 16 + mrb + v) * N + n0 + l15] = acc1[v];
        Y[(long)(m0 + 32 + mrb + v) * N + n0 + l15] = acc2[v];
        Y[(long)(m0 + 48 + mrb + v) * N + n0 + l15] = acc3[v];
    }
}

// ---------------------------------------------------------------------------
// LSTM recurrence (persistent grid, LDS-resident W_hh slice staged via
// CDNA5 async global->LDS, per-step grid sync through a monotonically
// increasing global atomic counter).
//   XP  : [T, 2000] precomputed x@W_ih^T + b_ih + b_hh
//   Whh : [2000, 500]
//   Hout: [T, 500]  (also serves as the h_{t-1} source for all blocks)
// Block b owns hidden units [b*UPB, b*UPB+UPB) => 20 gate rows.
// ---------------------------------------------------------------------------
__global__ void __launch_bounds__(128, 1)
lstm_scan_kernel(const float* __restrict__ XP,
                 const float* __restrict__ Whh,
                 float* __restrict__ Hout,
                 unsigned* __restrict__ ctr,
                 int T) {
    __shared__ float Wl[ROWS][HIDDEN];   // 40 KB
    __shared__ float h_l[HIDDEN];
    __shared__ float gates_l[ROWS];

    const int tid = threadIdx.x;
    const int u0  = blockIdx.x * UPB;

    // stage this block's W_hh slice into LDS once: 2500 x b128 async copies
    {
        const int V4 = HIDDEN / 4;                 // 125 float4 per row
        for (int idx = tid; idx < ROWS * V4; idx += blockDim.x) {
            const int r = idx / V4, q = idx - r * V4;
            const int gate = r / UPB, ul = r - gate * UPB;
            const float* g = Whh + (long)(gate * HIDDEN + u0 + ul) * HIDDEN + q * 4;
            async_g2l_b128(g, &Wl[r][q * 4]);
        }
        wait_async0();
    }

    const int wave = tid >> 5, lane = tid & 31;
    float c = 0.0f;   // persistent cell state (threads tid < UPB)

    for (int t = 0; t < T; ++t) {
        // previous h into LDS (t==0 -> zeros); async copy bypasses VGPRs
        if (t == 0) {
            for (int k = tid; k < HIDDEN; k += blockDim.x)
                h_l[k] = 0.0f;
        } else {
            const float* hsrc = Hout + (long)(t - 1) * HIDDEN;
            for (int k = tid; k < HIDDEN; k += blockDim.x)
                async_g2l_b32(hsrc + k, &h_l[k]);
            wait_async0();
        }
        __syncthreads();

        // 20 recurrent dot products: 5 rows per wave, 500-wide, shuffle-reduced
        for (int i = 0; i < UPB; ++i) {
            const int r = wave * UPB + i;
            float acc = 0.0f;
            for (int k = lane; k < HIDDEN; k += 32)
                acc += Wl[r][k] * h_l[k];
            for (int off = 16; off > 0; off >>= 1)
                acc += __shfl_down(acc, off, 32);
            if (lane == 0) gates_l[r] = acc;
        }
        __syncthreads();

        // gate nonlinearities + state update (PyTorch order i, f, g, o)
        if (tid < UPB) {
            const int u = u0 + tid;
            const float* xp = XP + (long)t * GATES;
            float gi = xp[0 * HIDDEN + u] + gates_l[0 * UPB + tid];
            float gf = xp[1 * HIDDEN + u] + gates_l[1 * UPB + tid];
            float gg = xp[2 * HIDDEN + u] + gates_l[2 * UPB + tid];
            float go = xp[3 * HIDDEN + u] + gates_l[3 * UPB + tid];
            gi = 1.0f / (1.0f + __expf(-gi));
            gf = 1.0f / (1.0f + __expf(-gf));
            gg = tanhf(gg);
            go = 1.0f / (1.0f + __expf(-go));
            c = gf * c + gi * gg;
            Hout[(long)t * HIDDEN + u] = go * tanhf(c);
        }

        // grid-wide barrier: release our h slice, wait for all NBLK blocks
        __threadfence();
        if (tid == 0) {
            __hip_atomic_fetch_add(ctr, 1u, __ATOMIC_RELEASE, __HIP_MEMORY_SCOPE_AGENT);
            const unsigned target = (unsigned)NBLK * (unsigned)(t + 1);
            while (__hip_atomic_load(ctr, __ATOMIC_ACQUIRE, __HIP_MEMORY_SCOPE_AGENT) < target) {
                __builtin_amdgcn_s_sleep(1);
            }
        }
        __syncthreads();
        __threadfence();   // acquire for all threads before reading Hout next step
    }
}

// ---------------------------------------------------------------------------
// final FC: out[cls] = h_last . fc_W[cls,:] + fc_b[cls]   (one wave per class)
// ---------------------------------------------------------------------------
__global__ void fc_kernel(const float* __restrict__ hlast,
                          const float* __restrict__ fcW,
                          const float* __restrict__ fcb,
                          float* __restrict__ out) {
    const int wave = threadIdx.x >> 5, lane = threadIdx.x & 31;
    if (wave >= NUM_CLASSES) return;
    float acc = 0.0f;
    const float* w = fcW + (long)wave * HIDDEN;
    for (int k = lane; k < HIDDEN; k += 32)
        acc += w[k] * hlast[k];
    for (int off = 16; off > 0; off >>= 1)
        acc += __shfl_down(acc, off, 32);
    if (lane == 0) out[wave] = acc + fcb[wave];
}

// ---------------------------------------------------------------------------
extern "C" void kernel_launch(void* const* d_in, const int* in_sizes, int n_in,
                              void* d_out, int out_size, void* d_ws, size_t ws_size,
                              hipStream_t stream) {
    const int*   seq   = (const int*)  d_in[0];
    const float* emb   = (const float*)d_in[1];
    const float* W_ih0 = (const float*)d_in[2];
    const float* W_hh0 = (const float*)d_in[3];
    const float* b_ih0 = (const float*)d_in[4];
    const float* b_hh0 = (const float*)d_in[5];
    const float* W_ih1 = (const float*)d_in[6];
    const float* W_hh1 = (const float*)d_in[7];
    const float* b_ih1 = (const float*)d_in[8];
    const float* b_hh1 = (const float*)d_in[9];
    const float* fc_W  = (const float*)d_in[10];
    const float* fc_b  = (const float*)d_in[11];
    float* out = (float*)d_out;

    // workspace layout (~43.4 MB of fp32 + 8 B counters)
    float* X   = (float*)d_ws;                        // [2048, 300]
    float* XP0 = X   + (size_t)SEQ_LEN * EMBED_DIM;   // [2048, 2000]
    float* H0  = XP0 + (size_t)SEQ_LEN * GATES;       // [2048, 500]
    float* XP1 = H0  + (size_t)SEQ_LEN * HIDDEN;      // [2048, 2000]
    float* H1  = XP1 + (size_t)SEQ_LEN * GATES;       // [2048, 500]
    unsigned* ctr = (unsigned*)(H1 + (size_t)SEQ_LEN * HIDDEN);

    init_ctr_kernel<<<1, 32, 0, stream>>>(ctr);

    // 1) embedding gather
    embed_kernel<<<SEQ_LEN, 128, 0, stream>>>(seq, emb, X);

    // 2) layer-0 input projection (fp32 WMMA GEMM): XP0 = X @ W_ih0^T + b
    inproj_wmma_kernel<<<dim3(GATES / 16, SEQ_LEN / 64), 32, 0, stream>>>(
        X, EMBED_DIM, W_ih0, b_ih0, b_hh0, XP0, GATES, EMBED_DIM);

    // 3) layer-0 recurrence
    lstm_scan_kernel<<<NBLK, 128, 0, stream>>>(XP0, W_hh0, H0, ctr + 0, SEQ_LEN);

    // 4) layer-1 input projection: XP1 = H0 @ W_ih1^T + b
    inproj_wmma_kernel<<<dim3(GATES / 16, SEQ_LEN / 64), 32, 0, stream>>>(
        H0, HIDDEN, W_ih1, b_ih1, b_hh1, XP1, GATES, HIDDEN);

    // 5) layer-1 recurrence
    lstm_scan_kernel<<<NBLK, 128, 0, stream>>>(XP1, W_hh1, H1, ctr + 1, SEQ_LEN);

    // 6) final FC on last timestep
    fc_kernel<<<1, 32 * NUM_CLASSES, 0, stream>>>(
        H1 + (size_t)(SEQ_LEN - 1) * HIDDEN, fc_W, fc_b, out);
}